// NodeNet_89137751261522
// MI455X (gfx1250) — compile-verified
//
#include <hip/hip_runtime.h>
#include <math.h>

// ---------------- problem constants (from reference) ----------------
#define N_NODES 100000
#define N_EDGES 1600000
#define HID 64
#define FD 67            // HID + 3
#define FDP 68           // FD zero-padded to multiple of 4
#define K1P (3 * FDP)    // 204: padded K of GEMM1

typedef __attribute__((ext_vector_type(2))) float v2f;
typedef __attribute__((ext_vector_type(8))) float v8f;

// -------------------------------------------------------------------
// Kernel 1: zero the mi/mo accumulators in workspace (float4 stores).
// -------------------------------------------------------------------
__global__ void nodenet_zero_kernel(float4* __restrict__ ws, int n4) {
    int i = blockIdx.x * blockDim.x + threadIdx.x;
    if (i < n4) ws[i] = make_float4(0.f, 0.f, 0.f, 0.f);
}

// -------------------------------------------------------------------
// Kernel 1b: transpose + zero-pad weights into workspace.
//   W1t[c][s*68 + k] = (k < 67) ? W1[(s*67+k)*64 + c] : 0   (64 x 204)
//   W2t[c][k]        = W2[k*64 + c]                          (64 x 64)
// Padding on the B side makes the GEMM1 K-loop branch-free, and since
// (k0 + 2*half) is always even, B fragments become aligned b64 loads.
// -------------------------------------------------------------------
__global__ void nodenet_wt_kernel(const float* __restrict__ W1,
                                  const float* __restrict__ W2,
                                  float* __restrict__ W1t,
                                  float* __restrict__ W2t) {
    int i = blockIdx.x * blockDim.x + threadIdx.x;
    if (i < HID * K1P) {                       // 13056
        int c = i / K1P, t = i - c * K1P;
        int s = t / FDP, k = t - s * FDP;
        W1t[i] = (k < FD) ? W1[(s * FD + k) * HID + c] : 0.0f;
    } else if (i < HID * K1P + HID * HID) {    // + 4096
        int j = i - HID * K1P;
        int c = j / HID, k = j - c * HID;
        W2t[j] = W2[k * HID + c];
    }
}

// -------------------------------------------------------------------
// Kernel 2: edge scatter.  One wave32 per edge; lanes stride the 67
// features.  mi[r] += w * X[s],  mo[s] += w * X[r].
// Working set (~80 MB) is L2-resident on MI455X (192 MB L2), so this
// is bound by L2 atomic throughput -> hardware global_atomic_add_f32.
// -------------------------------------------------------------------
__global__ void nodenet_edge_kernel(const float* __restrict__ X,
                                    const float* __restrict__ e,
                                    const int*   __restrict__ senders,
                                    const int*   __restrict__ receivers,
                                    float* __restrict__ mi,
                                    float* __restrict__ mo) {
    const int wavesPerBlock = blockDim.x >> 5;
    const int edge = blockIdx.x * wavesPerBlock + (threadIdx.x >> 5);
    if (edge >= N_EDGES) return;
    const int lane = threadIdx.x & 31;

    const int s = senders[edge];
    const int r = receivers[edge];
    const float w = e[edge];

    const float* __restrict__ xs = X + (size_t)s * FD;
    const float* __restrict__ xr = X + (size_t)r * FD;
    float* __restrict__ mir = mi + (size_t)r * FD;
    float* __restrict__ mos = mo + (size_t)s * FD;

    for (int f = lane; f < FD; f += 32) {
#if defined(__HIP_PLATFORM_AMD__)
        unsafeAtomicAdd(&mir[f], w * xs[f]);
        unsafeAtomicAdd(&mos[f], w * xr[f]);
#else
        atomicAdd(&mir[f], w * xs[f]);
        atomicAdd(&mos[f], w * xr[f]);
#endif
    }
}

// -------------------------------------------------------------------
// Kernel 3: fused MLP, exact fp32 via V_WMMA_F32_16X16X4_F32.
//   out = tanh( tanh( [mi|mo|X] @ W1 + b1 ) @ W2 + b2 )
// One wave owns TWO 16-row tiles (M=32): each B fragment feeds two
// wmma ops, doubling matrix-op density per load.  h is bounced
// through a wave-private LDS slab (C-layout -> A-layout).
// -------------------------------------------------------------------
#define WAVES_PER_BLOCK 4
#define N_PAIRS (N_NODES / 32)   // 3125, exact

__device__ __forceinline__ v8f wmma_f32(v2f a, v2f b, v8f c) {
    return __builtin_amdgcn_wmma_f32_16x16x4_f32(
        false, a, false, b, (short)0, c, false, false);
}

__global__ void nodenet_mlp_kernel(const float* __restrict__ X,
                                   const float* __restrict__ mi,
                                   const float* __restrict__ mo,
                                   const float* __restrict__ W1t,
                                   const float* __restrict__ b1,
                                   const float* __restrict__ W2t,
                                   const float* __restrict__ b2,
                                   float* __restrict__ out) {
    __shared__ float sH[WAVES_PER_BLOCK * 32 * HID];   // 32 KB

    const int wave = threadIdx.x >> 5;
    const int lane = threadIdx.x & 31;
    const int pair = blockIdx.x * WAVES_PER_BLOCK + wave;
    if (pair >= N_PAIRS) return;

    const int half  = lane >> 4;        // 0: lanes 0-15, 1: lanes 16-31
    const int laneM = lane & 15;
    const int kk    = 2 * half;         // K offset held by this half-wave
    const int rowBase = pair * 32;

    v8f acc[2][4];
    #pragma unroll
    for (int t = 0; t < 2; ++t)
        #pragma unroll
        for (int c = 0; c < 4; ++c) acc[t][c] = (v8f)0.0f;

    // ---- GEMM1: K = 201 (padded 204), three 67-wide sources -------
    const float* srcs[3] = { mi, mo, X };
    #pragma unroll
    for (int s = 0; s < 3; ++s) {
        const float* __restrict__ a0 = srcs[s] + (size_t)(rowBase + laneM) * FD;
        const float* __restrict__ a1 = a0 + 16 * FD;
        const float* __restrict__ bt = W1t + s * FDP;

        // main loop: K 0..63, branch-free (f0 = k0+kk <= 62, f1 <= 63)
        for (int k0 = 0; k0 < 64; k0 += 4) {
            const int f0 = k0 + kk;
            v2f A0, A1;
            A0.x = a0[f0]; A0.y = a0[f0 + 1];
            A1.x = a1[f0]; A1.y = a1[f0 + 1];
            #pragma unroll
            for (int c = 0; c < 4; ++c) {
                // (c*16+laneM)*204 + s*68 + f0 is even -> aligned b64
                v2f B = *(const v2f*)(bt + (c * 16 + laneM) * K1P + f0);
                acc[0][c] = wmma_f32(A0, B, acc[0][c]);
                acc[1][c] = wmma_f32(A1, B, acc[1][c]);
            }
        }
        // peeled step: K 64..67 (A pads K=67 with 0; B pad row is 0 too)
        {
            const int f0 = 64 + kk;                    // 64 or 66
            v2f A0, A1;
            A0.x = a0[f0]; A1.x = a1[f0];
            A0.y = (half == 0) ? a0[f0 + 1] : 0.0f;    // K=65 valid, K=67 pad
            A1.y = (half == 0) ? a1[f0 + 1] : 0.0f;
            #pragma unroll
            for (int c = 0; c < 4; ++c) {
                v2f B = *(const v2f*)(bt + (c * 16 + laneM) * K1P + f0);
                acc[0][c] = wmma_f32(A0, B, acc[0][c]);
                acc[1][c] = wmma_f32(A1, B, acc[1][c]);
            }
        }
    }

    // ---- bias + tanh, stage h into wave-private LDS (row major) ---
    float* __restrict__ hSlab = sH + wave * 32 * HID;
    #pragma unroll
    for (int t = 0; t < 2; ++t)
        #pragma unroll
        for (int c = 0; c < 4; ++c) {
            const int col = c * 16 + laneM;
            const float bias = b1[col];
            #pragma unroll
            for (int i = 0; i < 8; ++i) {
                const int r = t * 16 + i + 8 * half;
                hSlab[r * HID + col] = tanhf(acc[t][c][i] + bias);
            }
        }
    // wave-private region; DS ops from one wave stay in order.

    // ---- GEMM2: 64 -> 64, branch-free, aligned b64 A and B --------
    v8f acc2[2][4];
    #pragma unroll
    for (int t = 0; t < 2; ++t)
        #pragma unroll
        for (int c = 0; c < 4; ++c) acc2[t][c] = (v8f)0.0f;

    const float* __restrict__ h0 = hSlab + laneM * HID;
    const float* __restrict__ h1 = hSlab + (16 + laneM) * HID;
    for (int k0 = 0; k0 < HID; k0 += 4) {              // 16 steps
        const int f0 = k0 + kk;                        // even -> aligned
        v2f A0 = *(const v2f*)(h0 + f0);
        v2f A1 = *(const v2f*)(h1 + f0);
        #pragma unroll
        for (int c = 0; c < 4; ++c) {
            v2f B = *(const v2f*)(W2t + (c * 16 + laneM) * HID + f0);
            acc2[0][c] = wmma_f32(A0, B, acc2[0][c]);
            acc2[1][c] = wmma_f32(A1, B, acc2[1][c]);
        }
    }

    // ---- bias + tanh + store --------------------------------------
    #pragma unroll
    for (int t = 0; t < 2; ++t)
        #pragma unroll
        for (int c = 0; c < 4; ++c) {
            const int col = c * 16 + laneM;
            const float bias = b2[col];
            #pragma unroll
            for (int i = 0; i < 8; ++i) {
                const int r = t * 16 + i + 8 * half;
                out[(size_t)(rowBase + r) * HID + col] = tanhf(acc2[t][c][i] + bias);
            }
        }
}

// -------------------------------------------------------------------
// Host launcher
// -------------------------------------------------------------------
extern "C" void kernel_launch(void* const* d_in, const int* in_sizes, int n_in,
                              void* d_out, int out_size, void* d_ws, size_t ws_size,
                              hipStream_t stream) {
    const float* X        = (const float*)d_in[0];
    const float* e        = (const float*)d_in[1];
    const float* W1       = (const float*)d_in[2];
    const float* b1       = (const float*)d_in[3];
    const float* W2       = (const float*)d_in[4];
    const float* b2       = (const float*)d_in[5];
    const int*   senders  = (const int*)d_in[6];
    const int*   receivers= (const int*)d_in[7];
    float* out = (float*)d_out;

    // workspace layout (floats): mi | mo | W1t | W2t
    float* mi  = (float*)d_ws;                       // N_NODES*FD
    float* mo  = mi  + (size_t)N_NODES * FD;         // N_NODES*FD
    float* W1t = mo  + (size_t)N_NODES * FD;         // HID*K1P = 13056
    float* W2t = W1t + (size_t)HID * K1P;            // HID*HID = 4096

    // 1) zero accumulators (re-zeroed every call -> deterministic)
    {
        const int n4 = (2 * N_NODES * FD) / 4;       // 13,400,000 / 4
        nodenet_zero_kernel<<<(n4 + 255) / 256, 256, 0, stream>>>(
            (float4*)d_ws, n4);
    }
    // 1b) transpose + pad weights
    {
        const int n = HID * K1P + HID * HID;         // 17152
        nodenet_wt_kernel<<<(n + 255) / 256, 256, 0, stream>>>(W1, W2, W1t, W2t);
    }
    // 2) edge scatter: one wave per edge, 8 waves per 256-thread block
    {
        const int blk = 256;
        const int wavesPerBlock = blk / 32;
        const int grid = (N_EDGES + wavesPerBlock - 1) / wavesPerBlock;
        nodenet_edge_kernel<<<grid, blk, 0, stream>>>(
            X, e, senders, receivers, mi, mo);
    }
    // 3) fused WMMA MLP: one wave per 32-row pair of tiles
    {
        const int blk = 32 * WAVES_PER_BLOCK;
        const int grid = (N_PAIRS + WAVES_PER_BLOCK - 1) / WAVES_PER_BLOCK;
        nodenet_mlp_kernel<<<grid, blk, 0, stream>>>(
            X, mi, mo, W1t, b1, W2t, b2, out);
    }
    (void)in_sizes; (void)n_in; (void)out_size; (void)ws_size;
}